// FrequencySelectiveDilateLoss_34634616275210
// MI455X (gfx1250) — compile-verified
//
#include <hip/hip_runtime.h>
#include <math.h>

#define NN       256
#define BATCH    128
#define FDIM     4
#define NMAT     (BATCH * FDIM)        // 512 softDTW matrices
#define NELEM    (BATCH * NN * FDIM)   // 131072 tensor elements
#define BIGF     1.0e10f
#define GAMMA_F  0.01f
#define INVG     100.0f

typedef __attribute__((ext_vector_type(2))) float v2f;
typedef __attribute__((ext_vector_type(8))) float v8f;

// ---------------------------------------------------------------------------
// K1: build circulant high-pass projection matrix H (256x256).
// H[t,s] = (1/256) * ( 2*sum_{f=103..127} cos(2*pi*f*(t-s)/256) + (-1)^(t-s) )
// ---------------------------------------------------------------------------
__global__ void build_h_kernel(float* __restrict__ H) {
    const int t = blockIdx.x;
    const int s = threadIdx.x;
    const int delta = t - s;
    double acc = (delta & 1) ? -1.0 : 1.0;      // Nyquist f=128 term
    const double base = 2.0 * 3.14159265358979323846 * (double)delta / 256.0;
    for (int fr = 103; fr <= 127; ++fr)
        acc += 2.0 * cos(base * (double)fr);
    H[t * NN + s] = (float)(acc / 256.0);
}

// ---------------------------------------------------------------------------
// K2: Y = X * H^T via V_WMMA_F32_16X16X4_F32.
// Rows r = b*4 + f (512), cols = t (256), K = s (256).
// X layout is [b, s, f]; output Y written back in the same [b, t, f] layout.
// One wave (32 threads) per 16x16 output tile; blockIdx.z picks tensor.
// ---------------------------------------------------------------------------
__global__ void project_wmma_kernel(const float* __restrict__ O,
                                    const float* __restrict__ T,
                                    const float* __restrict__ H,
                                    float* __restrict__ OH,
                                    float* __restrict__ TH) {
    const float* X = (blockIdx.z == 0) ? O : T;
    float*       Y = (blockIdx.z == 0) ? OH : TH;

    const int tm   = blockIdx.x;        // 0..31 (M tiles over 512 rows)
    const int tn   = blockIdx.y;        // 0..15 (N tiles over 256 cols)
    const int lane = threadIdx.x;       // 0..31
    const int ml   = lane & 15;         // M (for A) / N (for B) within tile
    const int kh   = (lane >> 4) << 1;  // K half: lanes 0-15 -> K{0,1}, 16-31 -> K{2,3}

    const int r  = tm * 16 + ml;        // global row  (= b*4 + f)
    const int bb = r >> 2, ff = r & 3;
    const int nn = tn * 16 + ml;        // global col  (= t)

    v8f c = {0.f, 0.f, 0.f, 0.f, 0.f, 0.f, 0.f, 0.f};
    for (int k0 = 0; k0 < NN; k0 += 4) {
        v2f a, b;
        // A[m, k] = X[bb, k, ff]
        a.x = X[bb * (NN * FDIM) + (k0 + kh)     * FDIM + ff];
        a.y = X[bb * (NN * FDIM) + (k0 + kh + 1) * FDIM + ff];
        // B[k, n] = H^T[k, n] = H[n, k]   (H is symmetric anyway)
        b.x = H[nn * NN + k0 + kh];
        b.y = H[nn * NN + k0 + kh + 1];
        c = __builtin_amdgcn_wmma_f32_16x16x4_f32(
                /*neg_a=*/false, a, /*neg_b=*/false, b,
                /*c_mod=*/(short)0, c, /*reuse_a=*/false, /*reuse_b=*/false);
    }
    // C/D layout: M = v + 8*(lane>=16), N = lane&15
    const int rbase = tm * 16 + ((lane >> 4) << 3);
    const int col   = tn * 16 + ml;
#pragma unroll
    for (int v = 0; v < 8; ++v) {
        const int row = rbase + v;
        Y[(row >> 2) * (NN * FDIM) + col * FDIM + (row & 3)] = c[v];
    }
}

// ---------------------------------------------------------------------------
// K3: per-block partial sums of he = sum(high_o^2), le = sum(low_o^2),
//     low_l1 = sum(|low_o - low_t|), with low = x - high.
// ---------------------------------------------------------------------------
__global__ void energies_kernel(const float* __restrict__ O,
                                const float* __restrict__ T,
                                const float* __restrict__ OH,
                                const float* __restrict__ TH,
                                float* __restrict__ he_p,
                                float* __restrict__ le_p,
                                float* __restrict__ low_p) {
    const int tid = threadIdx.x;
    const int gid = blockIdx.x * blockDim.x + tid;
    const int stride = gridDim.x * blockDim.x;
    float he = 0.f, le = 0.f, lo = 0.f;
    for (int i = gid; i < NELEM; i += stride) {
        const float oh = OH[i];
        const float ol = O[i] - oh;
        const float tl = T[i] - TH[i];
        he += oh * oh;
        le += ol * ol;
        lo += fabsf(ol - tl);
    }
    __shared__ float s1[256], s2[256], s3[256];
    s1[tid] = he; s2[tid] = le; s3[tid] = lo;
    __syncthreads();
    for (int s = 128; s > 0; s >>= 1) {
        if (tid < s) { s1[tid] += s1[tid + s]; s2[tid] += s2[tid + s]; s3[tid] += s3[tid + s]; }
        __syncthreads();
    }
    if (tid == 0) { he_p[blockIdx.x] = s1[0]; le_p[blockIdx.x] = s2[0]; low_p[blockIdx.x] = s3[0]; }
}

// ---------------------------------------------------------------------------
// K4: soft-DTW forward value + analytic gradient, one 256x256 matrix per
// workgroup. Full R matrix (256 KB) lives in LDS (320 KB/WGP on CDNA5).
// Forward: R[i,j] = D[i,j] + softmin_g(up, left, diag) over anti-diagonals.
// Backward: E[i,j] = sum_nbr E[a,b]*exp((R[a,b]-D[a,b]-R[i,j])/g); exponent<=0.
// Accumulates sum(E * (i-j)^2) on the fly (Omega term).
// ---------------------------------------------------------------------------
__global__ void softdtw_kernel(const float* __restrict__ OH,
                               const float* __restrict__ TH,
                               float* __restrict__ sdtw_vals,
                               float* __restrict__ eomega) {
    const int g   = blockIdx.x;     // 0..511
    const int f   = g & 3;
    const int b   = g >> 2;
    const int tid = threadIdx.x;    // 0..255 == row index i

    extern __shared__ float lds[];
    float* R   = lds;               // 65536 floats
    float* tv  = lds + NN * NN;     // 256
    float* ov  = tv + NN;           // 256
    float* eA  = ov + NN;           // 256  (E at diag d+2)
    float* eB  = eA + NN;           // 256  (E at diag d+1)
    float* eC  = eB + NN;           // 256  (E at diag d, current)
    float* red = eC + NN;           // 256  (block reduction)

    tv[tid] = TH[b * (NN * FDIM) + tid * FDIM + f];
    ov[tid] = OH[b * (NN * FDIM) + tid * FDIM + f];
    __syncthreads();
    const float ti = tv[tid];

    // ---- forward anti-diagonal scan ----
    for (int d = 0; d < 2 * NN - 1; ++d) {
        const int j = d - tid;
        if (j >= 0 && j < NN) {
            const float diff = ti - ov[j];
            const float Dij  = diff * diff;
            float rv;
            if (d == 0) {
                rv = Dij;
            } else {
                const float up = (tid >= 1)           ? R[(tid - 1) * NN + j]     : BIGF;
                const float lf = (j >= 1)             ? R[tid * NN + j - 1]       : BIGF;
                const float dg = (tid >= 1 && j >= 1) ? R[(tid - 1) * NN + j - 1] : BIGF;
                const float m  = fminf(up, fminf(lf, dg));
                const float s  = __expf((m - up) * INVG) + __expf((m - lf) * INVG)
                               + __expf((m - dg) * INVG);
                rv = Dij + m - GAMMA_F * __logf(s);
            }
            R[tid * NN + j] = rv;
        }
        __syncthreads();
    }

    // ---- backward gradient scan (rolling E diagonals) ----
    eA[tid] = 0.f;
    eB[tid] = 0.f;
    __syncthreads();
    float* ed2 = eA;   // diag d+2
    float* ed1 = eB;   // diag d+1
    float* ecu = eC;   // diag d
    float acc = 0.f;
    for (int d = 2 * NN - 2; d >= 0; --d) {
        const int j = d - tid;
        float E = 0.f;
        if (j >= 0 && j < NN) {
            if (d == 2 * NN - 2) {
                E = 1.0f;
            } else {
                const float Rij = R[tid * NN + j];
                if (j + 1 < NN) {                       // neighbor (i, j+1)
                    const float df = ti - ov[j + 1];
                    E += ed1[tid] * __expf((R[tid * NN + j + 1] - df * df - Rij) * INVG);
                }
                if (tid + 1 < NN) {                     // neighbor (i+1, j)
                    const float df = tv[tid + 1] - ov[j];
                    E += ed1[tid + 1] * __expf((R[(tid + 1) * NN + j] - df * df - Rij) * INVG);
                }
                if (tid + 1 < NN && j + 1 < NN) {       // neighbor (i+1, j+1)
                    const float df = tv[tid + 1] - ov[j + 1];
                    E += ed2[tid + 1] * __expf((R[(tid + 1) * NN + j + 1] - df * df - Rij) * INVG);
                }
            }
            const float om = (float)(tid - j);          // Omega = (i - j)^2
            acc += E * om * om;
        }
        ecu[tid] = E;                                   // invalid rows store 0
        __syncthreads();
        float* t0 = ed2; ed2 = ed1; ed1 = ecu; ecu = t0;
    }

    red[tid] = acc;
    __syncthreads();
    for (int s = 128; s > 0; s >>= 1) {
        if (tid < s) red[tid] += red[tid + s];
        __syncthreads();
    }
    if (tid == 0) {
        sdtw_vals[g] = R[(NN - 1) * NN + (NN - 1)];
        eomega[g]    = red[0];
    }
}

// ---------------------------------------------------------------------------
// K5: deterministic final reduction + scalar combination -> 6 outputs.
// ---------------------------------------------------------------------------
__global__ void finalize_kernel(const float* __restrict__ sv,
                                const float* __restrict__ eo,
                                const float* __restrict__ hp,
                                const float* __restrict__ lp,
                                const float* __restrict__ wp,
                                float* __restrict__ out) {
    __shared__ float s[256];
    const int tid = threadIdx.x;
    float sums[5];
    const float* srcs[5] = {sv, eo, hp, lp, wp};
    const int    cnts[5] = {NMAT, NMAT, 128, 128, 128};
#pragma unroll 1
    for (int k = 0; k < 5; ++k) {
        float a = 0.f;
        for (int i = tid; i < cnts[k]; i += 256) a += srcs[k][i];
        s[tid] = a;
        __syncthreads();
        for (int t = 128; t > 0; t >>= 1) {
            if (tid < t) s[tid] += s[tid + t];
            __syncthreads();
        }
        sums[k] = s[0];
        __syncthreads();
    }
    if (tid == 0) {
        const float shape    = sums[0] / (float)NMAT;
        const float temporal = sums[1] / ((float)NMAT * (float)(NN * NN));
        const float high     = 0.5f * shape + 0.5f * temporal;   // ALPHA = 0.5
        const float he       = sums[2] / (float)NELEM;
        const float le       = sums[3] / (float)NELEM;
        const float low      = sums[4] / (float)NELEM;
        const float tot      = he + le + 1e-8f;
        const float total    = le / tot * low + he / tot * high;
        out[0] = total;
        out[1] = low;
        out[2] = high;
        out[3] = shape;
        out[4] = temporal;
        out[5] = 26.0f / 129.0f;   // high_freq_ratio: freqs 103..128 of 129
    }
}

// ---------------------------------------------------------------------------
extern "C" void kernel_launch(void* const* d_in, const int* in_sizes, int n_in,
                              void* d_out, int out_size, void* d_ws, size_t ws_size,
                              hipStream_t stream) {
    const float* O = (const float*)d_in[0];   // outputs (128,256,4) f32
    const float* T = (const float*)d_in[1];   // targets (128,256,4) f32
    float* ws = (float*)d_ws;

    float* H     = ws;                   // 65536
    float* OH    = H + NN * NN;          // 131072
    float* TH    = OH + NELEM;           // 131072
    float* sdtw  = TH + NELEM;           // 512
    float* eom   = sdtw + NMAT;          // 512
    float* he_p  = eom + NMAT;           // 128
    float* le_p  = he_p + 128;           // 128
    float* low_p = le_p + 128;           // 128
    float* out   = (float*)d_out;        // 6 scalars

    build_h_kernel<<<NN, NN, 0, stream>>>(H);

    {
        dim3 grid(NMAT / 16, NN / 16, 2);   // 32 x 16 x 2 tiles
        dim3 block(32);                     // one wave per tile (wave32)
        project_wmma_kernel<<<grid, block, 0, stream>>>(O, T, H, OH, TH);
    }

    energies_kernel<<<128, 256, 0, stream>>>(O, T, OH, TH, he_p, le_p, low_p);

    const size_t lds_bytes = (size_t)(NN * NN + 6 * NN) * sizeof(float); // 268288 B < 320KB
    (void)hipFuncSetAttribute((const void*)softdtw_kernel,
                              hipFuncAttributeMaxDynamicSharedMemorySize,
                              (int)lds_bytes);
    softdtw_kernel<<<NMAT, NN, lds_bytes, stream>>>(OH, TH, sdtw, eom);

    finalize_kernel<<<1, 256, 0, stream>>>(sdtw, eom, he_p, le_p, low_p, out);
}